// Tower_79010218377313
// MI455X (gfx1250) — compile-verified
//
#include <hip/hip_runtime.h>
#include <hip/hip_bf16.h>
#include <math.h>

// ---------------------------------------------------------------------------
// MI455X / gfx1250 plan
//   Problem: emb gather -> inp_proj GEMM (6.7 GF) -> 200-step tanh RNN scan
//            (6.7 GF, sequential) -> L2 normalize.  All fp32.
//   emb table (51 MB) and inp_proj workspace (105 MB) both fit the 192 MB L2,
//   so the scan is latency-bound, not bandwidth-bound.  Both GEMMs use the
//   CDNA5 fp32 matrix op V_WMMA_F32_16X16X4_F32 (exact precision match),
//   formulated as D = W x h^T so W lives in VGPRs (loop invariant A operand).
//   Scan critical path per step: 8-deep WMMA chain (4 parallel chains) +
//   hardware V_TANH_F32 + one LDS round trip + one workgroup barrier.
// ---------------------------------------------------------------------------

typedef float v2f __attribute__((ext_vector_type(2)));
typedef float v8f __attribute__((ext_vector_type(8)));

#define BATCH  1024
#define SEQ    200
#define EMBED  128
#define NBT    64      // batch tiles of 16 rows

static __device__ __forceinline__ v8f wmma_f32x4(v2f a, v2f b, v8f c) {
  // 8 args: (neg_a, A, neg_b, B, c_mod, C, reuse_a, reuse_b)
  return __builtin_amdgcn_wmma_f32_16x16x4_f32(false, a, false, b, (short)0, c,
                                               false, false);
}

// Hardware transcendental tanh (TRANS32 op on gfx1250).  Prefer the builtin
// (compiler models the TRANS data hazard); fall back to inline asm with a
// trailing independent VALU op per the "1 indep op after TRANS" rule.
static __device__ __forceinline__ float fast_tanh(float x) {
#if __has_builtin(__builtin_amdgcn_tanhf)
  return __builtin_amdgcn_tanhf(x);
#else
  float y;
  asm volatile("v_tanh_f32 %0, %1\n\tv_nop\n\tv_nop" : "=v"(y) : "v"(x));
  return y;
#endif
}

// ---------------------------------------------------------------------------
// Kernel 1: fused embedding gather + input projection (+ b_in), all (b,s).
//   Block = (s, bt): 16 gathered emb rows -> LDS, then 8 waves each compute a
//   16(f) x 16(n) tile of ip^T = W_in x emb^T via 32 fp32 WMMAs.
//   Output layout ws[((s*64+bt)*128 + f)*16 + n]  (ip^T tiles, D-layout rows).
// ---------------------------------------------------------------------------
__global__ __launch_bounds__(256) void rnn_proj_kernel(
    const int* __restrict__ x, const float* __restrict__ emb,
    const float* __restrict__ Win, const float* __restrict__ bin,
    float* __restrict__ ws) {
  const int tid  = threadIdx.x;
  const int wave = tid >> 5;     // 0..7  -> f tile
  const int lane = tid & 31;
  const int half = lane >> 4;    // 0/1
  const int l16  = lane & 15;
  const int s    = blockIdx.x >> 6;   // 0..199
  const int bt   = blockIdx.x & 63;   // 0..63

  __shared__ float embLDS[16][EMBED];

  // Cooperative gather: 16 rows x 128 floats, 8 floats per thread (coalesced
  // 32B chunks; 16 threads cover one 512B row).  padding_idx=0 row is already
  // zero in the table.
  {
    const int row  = tid >> 4;         // 0..15
    const int col8 = (tid & 15) * 8;   // 0..120
    const int idx  = x[(bt * 16 + row) * SEQ + s];
    const float* src = emb + (size_t)idx * EMBED + col8;
    float4 p0 = *(const float4*)(src);
    float4 p1 = *(const float4*)(src + 4);
    *(float4*)&embLDS[row][col8]     = p0;
    *(float4*)&embLDS[row][col8 + 4] = p1;
  }

  // A operand = W_in f-tile (M=f=16, K=e=128): fp32 A 16x4 layout per ISA:
  // lanes 0-15 hold K = k0,k0+1 ; lanes 16-31 hold K = k0+2,k0+3.
  const int fRow = wave * 16 + l16;
  v2f A[32];
#pragma unroll
  for (int k = 0; k < 32; ++k)
    A[k] = *(const v2f*)(Win + (size_t)fRow * EMBED + k * 4 + half * 2);

  __syncthreads();

  // Accumulators: C/D layout: VGPR r holds M = r (lanes 0-15) / r+8 (16-31).
  v8f acc0, acc1;
#pragma unroll
  for (int r = 0; r < 8; ++r) {
    acc0[r] = bin[wave * 16 + r + half * 8];  // fold b_in into C
    acc1[r] = 0.0f;
  }
  // Two independent K-chains (throughput-bound kernel; depth is fine).
#pragma unroll
  for (int k = 0; k < 16; ++k) {
    v2f b = *(const v2f*)&embLDS[l16][k * 4 + half * 2];  // B[k][n]=emb[n][k]
    acc0 = wmma_f32x4(A[k], b, acc0);
  }
#pragma unroll
  for (int k = 16; k < 32; ++k) {
    v2f b = *(const v2f*)&embLDS[l16][k * 4 + half * 2];
    acc1 = wmma_f32x4(A[k], b, acc1);
  }
#pragma unroll
  for (int r = 0; r < 8; ++r) acc0[r] += acc1[r];

  // Store ip^T tile.
  float* dst = ws + ((size_t)(s * NBT + bt) * EMBED + wave * 16 + half * 8) * 16 + l16;
#pragma unroll
  for (int r = 0; r < 8; ++r) dst[r * 16] = acc0[r];
}

// ---------------------------------------------------------------------------
// Kernel 2: recurrent scan + L2 normalize.  One block per batch tile (64
//   blocks x 8 waves).  Wave w owns the 16-row f-tile of W_h (A operand,
//   preloaded: 64 VGPRs).  h^T lives in LDS, double-buffered, stored
//   pair-interleaved so every B fragment is one 8-byte ds read (the compiler
//   fuses pairs into ds_load_2addr_b64):
//       hT[buf][e>>1][n*2 + (e&1)]
//   Per step: prefetch next ip tile, 32 WMMAs in 4 independent chains of 8,
//   +b_h, hardware tanh, write h^T, one barrier.
// ---------------------------------------------------------------------------
__global__ __launch_bounds__(256) void rnn_scan_kernel(
    const float* __restrict__ ws, const float* __restrict__ Wh,
    const float* __restrict__ bh, float* __restrict__ out) {
  const int tid  = threadIdx.x;
  const int wave = tid >> 5;
  const int lane = tid & 31;
  const int half = lane >> 4;
  const int l16  = lane & 15;
  const int bt   = blockIdx.x;  // 0..63

  __shared__ float hT[2][64][32];  // [buf][e>>1][n*2+(e&1)]  (2 x 8 KB)
  __shared__ float nrm[16];

  // h_{-1} = 0 lives in buffer 1 (step s reads buf[(s+1)&1], writes buf[s&1]).
  for (int i = tid; i < 64 * 32; i += 256) (&hT[1][0][0])[i] = 0.0f;
  if (tid < 16) nrm[tid] = 0.0f;

  // Preload W_h A fragments (loop invariant) and b_h fragment.
  const int fRow = wave * 16 + l16;
  v2f A[32];
#pragma unroll
  for (int k = 0; k < 32; ++k)
    A[k] = *(const v2f*)(Wh + (size_t)fRow * EMBED + k * 4 + half * 2);
  float bhf[8];
#pragma unroll
  for (int r = 0; r < 8; ++r) bhf[r] = bh[wave * 16 + r + half * 8];

  __syncthreads();

  const size_t sStride = (size_t)NBT * EMBED * 16;  // 131072 floats per step
  const float* ipBase =
      ws + ((size_t)bt * EMBED + wave * 16 + half * 8) * 16 + l16;

  float ip[8];
#pragma unroll
  for (int r = 0; r < 8; ++r) ip[r] = ipBase[r * 16];  // s = 0 tile

  v8f h;
  for (int s = 0; s < SEQ; ++s) {
    const int rb = (s + 1) & 1, wb = s & 1;

    // Register-prefetch the (independent) next ip tile.
    float ipn[8];
    if (s + 1 < SEQ) {
      const float* p = ipBase + (size_t)(s + 1) * sStride;
#pragma unroll
      for (int r = 0; r < 8; ++r) ipn[r] = p[r * 16];
    } else {
#pragma unroll
      for (int r = 0; r < 8; ++r) ipn[r] = 0.0f;
    }

    // Four independent 8-deep WMMA chains -> shortest serial latency.
    v8f acc0, acc1, acc2, acc3;
#pragma unroll
    for (int r = 0; r < 8; ++r) {
      acc0[r] = ip[r];
      acc1[r] = 0.0f;
      acc2[r] = 0.0f;
      acc3[r] = 0.0f;
    }
#pragma unroll
    for (int k = 0; k < 8; ++k) {
      v2f b = *(const v2f*)&hT[rb][2 * k + half][l16 * 2];
      acc0 = wmma_f32x4(A[k], b, acc0);
    }
#pragma unroll
    for (int k = 8; k < 16; ++k) {
      v2f b = *(const v2f*)&hT[rb][2 * k + half][l16 * 2];
      acc1 = wmma_f32x4(A[k], b, acc1);
    }
#pragma unroll
    for (int k = 16; k < 24; ++k) {
      v2f b = *(const v2f*)&hT[rb][2 * k + half][l16 * 2];
      acc2 = wmma_f32x4(A[k], b, acc2);
    }
#pragma unroll
    for (int k = 24; k < 32; ++k) {
      v2f b = *(const v2f*)&hT[rb][2 * k + half][l16 * 2];
      acc3 = wmma_f32x4(A[k], b, acc3);
    }

#pragma unroll
    for (int r = 0; r < 8; ++r)
      h[r] = fast_tanh((acc0[r] + acc1[r]) + (acc2[r] + acc3[r]) + bhf[r]);

    // Write h_new^T, pair-interleaved.
#pragma unroll
    for (int r = 0; r < 8; ++r) {
      const int e = wave * 16 + r + half * 8;
      hT[wb][e >> 1][l16 * 2 + (e & 1)] = h[r];
    }
#pragma unroll
    for (int r = 0; r < 8; ++r) ip[r] = ipn[r];

    __syncthreads();
  }

  // L2 normalize final h along E: reduce squares per batch row n = l16.
  float ss = 0.0f;
#pragma unroll
  for (int r = 0; r < 8; ++r) ss += h[r] * h[r];
  atomicAdd(&nrm[l16], ss);  // ds_add_f32
  __syncthreads();
  const float scale = 1.0f / fmaxf(sqrtf(nrm[l16]), 1e-12f);

  float* o = out + (size_t)(bt * 16 + l16) * EMBED + wave * 16 + half * 8;
#pragma unroll
  for (int r = 0; r < 8; ++r) o[r] = h[r] * scale;
}

// ---------------------------------------------------------------------------
extern "C" void kernel_launch(void* const* d_in, const int* in_sizes, int n_in,
                              void* d_out, int out_size, void* d_ws,
                              size_t ws_size, hipStream_t stream) {
  (void)in_sizes; (void)n_in; (void)out_size; (void)ws_size;
  const int*   x    = (const int*)d_in[0];
  const float* emb  = (const float*)d_in[1];
  const float* Win  = (const float*)d_in[2];
  const float* bin  = (const float*)d_in[3];
  const float* Wh   = (const float*)d_in[4];
  const float* bh   = (const float*)d_in[5];
  float*       ws   = (float*)d_ws;   // needs SEQ*NBT*128*16*4 = 104.9 MB
  float*       out  = (float*)d_out;  // [1024,128] fp32

  rnn_proj_kernel<<<SEQ * NBT, 256, 0, stream>>>(x, emb, Win, bin, ws);
  rnn_scan_kernel<<<NBT, 256, 0, stream>>>(ws, Wh, bh, out);
}